// BigNetwork_29798483099653
// MI455X (gfx1250) — compile-verified
//
#include <hip/hip_runtime.h>

typedef float v2f __attribute__((ext_vector_type(2)));
typedef float v8f __attribute__((ext_vector_type(8)));

#define K_HEADS        32
#define ROWS_PER_WAVE  16
#define ROWS_PER_BLOCK 128   // 8 waves * 16 rows
#define LDS_STRIDE     132   // 128 + 4 pad: 528B row (16B aligned), spreads banks

// out[k*N + n] = dot(x[n,:], W[k,0,:]) + b[k]
// One wave: 16 rows x 32 heads via two V_WMMA_F32_16X16X4_F32 (K=4 == feature dim).
// Output staged in LDS and written with GLOBAL_STORE_ASYNC_FROM_LDS_B128 (direct
// LDS->L2 DMA, ASYNCcnt-tracked) so the 256MB head-major output stream is fully
// coalesced 128-bit stores with no VGPR round-trip.
__global__ __launch_bounds__(256)
void linheads_wmma_kernel(const float* __restrict__ x,
                          const float* __restrict__ W,
                          const float* __restrict__ bias,
                          float* __restrict__ out,
                          int N) {
    __shared__ float tile[K_HEADS * LDS_STRIDE];

    const int tid      = threadIdx.x;
    const int lane     = tid & 31;
    const int wave     = tid >> 5;
    const int n0       = blockIdx.x * ROWS_PER_BLOCK;
    const int wave_row = wave * ROWS_PER_WAVE;

    // ---- A: 16x4 f32 tile of x ----
    // lanes 0-15: M=lane, VGPR0=K0, VGPR1=K1 ; lanes 16-31: M=lane-16, VGPR0=K2, VGPR1=K3
    const int m  = lane & 15;
    const int kk = (lane >> 4) * 2;          // feature pair base: 0 or 2
    int row = n0 + wave_row + m;
    row = (row < N) ? row : (N - 1);         // clamp keeps EXEC all-1s for WMMA
    v2f a = *(const v2f*)(x + (size_t)row * 4 + kk);

    // ---- B: W^T 4x16 tiles, heads 0-15 (b0) and 16-31 (b1) ----
    // mirrors A layout: lane L holds column N=L&15, K pair per lane half
    const int h = lane & 15;
    v2f b0 = *(const v2f*)(W + (size_t)h * 4 + kk);
    v2f b1 = *(const v2f*)(W + (size_t)(h + 16) * 4 + kk);

    // ---- C seeded with bias (C column == head) ----
    const float bv0 = bias[h];
    const float bv1 = bias[h + 16];
    v8f c0, c1;
#pragma unroll
    for (int j = 0; j < 8; ++j) { c0[j] = bv0; c1[j] = bv1; }

    // D = A x B + C  (8 args: neg_a, A, neg_b, B, c_mod, C, reuse_a, reuse_b)
    c0 = __builtin_amdgcn_wmma_f32_16x16x4_f32(false, a, false, b0, (short)0, c0, false, false);
    c1 = __builtin_amdgcn_wmma_f32_16x16x4_f32(false, a, false, b1, (short)0, c1, false, false);

    // ---- C layout -> LDS tile[head][local_n] ----
    // C 16x16 f32: N(col)=lane&15 (head), M(row)=j + 8*(lane>=16)
    const int hi = (lane >> 4) * 8;
#pragma unroll
    for (int j = 0; j < 8; ++j) {
        const int ln = wave_row + j + hi;                 // local n within block tile
        tile[h * LDS_STRIDE + ln]        = c0[j];
        tile[(h + 16) * LDS_STRIDE + ln] = c1[j];
    }

    __syncthreads();

    if (n0 + ROWS_PER_BLOCK <= N) {
        // ---- full block (all but possibly the last): async LDS->global B128 DMA ----
        // 32 heads x 128 n = 1024 x float4; 4 per thread; consecutive lanes cover
        // 512B-contiguous segments of one head row => fully coalesced.
#pragma unroll
        for (int it = 0; it < 4; ++it) {
            const int flat = it * 256 + tid;   // 0..1023
            const int head = flat >> 5;        // 32 float4 per head row
            const int i4   = (flat & 31) * 4;
            const unsigned lds_off =
                (unsigned)(uintptr_t)(tile + head * LDS_STRIDE + i4);
            const unsigned long long gaddr =
                (unsigned long long)(uintptr_t)(out + (size_t)head * N + n0 + i4);
            asm volatile("global_store_async_from_lds_b128 %0, %1, off"
                         :: "v"(gaddr), "v"(lds_off)
                         : "memory");
        }
        asm volatile("s_wait_asynccnt 0x0" ::: "memory");
    } else {
        // ---- cold tail path: scalar, bounds-checked ----
        for (int idx = tid; idx < K_HEADS * ROWS_PER_BLOCK; idx += 256) {
            const int head = idx >> 7;         // 128 n per head
            const int i    = idx & 127;
            const int n    = n0 + i;
            if (n < N) out[(size_t)head * N + n] = tile[head * LDS_STRIDE + i];
        }
    }
}

extern "C" void kernel_launch(void* const* d_in, const int* in_sizes, int n_in,
                              void* d_out, int out_size, void* d_ws, size_t ws_size,
                              hipStream_t stream) {
    const float* x    = (const float*)d_in[0];  // [N,4]
    const float* W    = (const float*)d_in[1];  // [32,1,4]
    const float* bias = (const float*)d_in[2];  // [32,1]
    float* out        = (float*)d_out;          // [32,N,1] flat

    const int N = in_sizes[0] / 4;
    const int blocks = (N + ROWS_PER_BLOCK - 1) / ROWS_PER_BLOCK;
    linheads_wmma_kernel<<<blocks, 256, 0, stream>>>(x, W, bias, out, N);
}